// DiceLoss_71305047048859
// MI455X (gfx1250) — compile-verified
//
#include <hip/hip_runtime.h>

// DiceLoss over 57-label volumes: 3-way histogram -> Dice -> scalar.
// Memory-bound: 134 MB traffic, ~5.8us at 23.3 TB/s HBM (L2-resident on replay).

#define NBINS        64   // padded from 57 -> every bin hits a distinct LDS bank
#define THREADS      256  // 8 wave32 per block
#define WAVES        8
#define GRID         2048

// ---------------------------------------------------------------- zero hist
__global__ void zero_hist_kernel(unsigned* __restrict__ ghist) {
    int t = threadIdx.x;
    if (t < 3 * NBINS) ghist[t] = 0u;
}

// ---------------------------------------------------------------- histogram
__global__ __launch_bounds__(THREADS)
void hist_kernel(const int* __restrict__ pred,
                 const int* __restrict__ tgt,
                 unsigned* __restrict__ ghist,
                 long long nchunks) {
    // Per-wave private histograms: wave32 -> 8 copies, zero cross-wave contention.
    __shared__ unsigned lh[WAVES][3 * NBINS];   // 6 KB of 320 KB LDS

    unsigned* flat = &lh[0][0];
    for (int i = threadIdx.x; i < WAVES * 3 * NBINS; i += THREADS)
        flat[i] = 0u;
    __syncthreads();

    const int wave = threadIdx.x >> 5;          // wave32 (gfx1250)
    unsigned* hp = lh[wave];                    // pred counts
    unsigned* ht = hp + NBINS;                  // true counts
    unsigned* hi = hp + 2 * NBINS;              // intersection counts

    const int4* p4 = (const int4*)pred;
    const int4* t4 = (const int4*)tgt;
    const long long stride = (long long)gridDim.x * THREADS;

    long long i = (long long)blockIdx.x * THREADS + threadIdx.x;

    // Unroll x2: 4 independent global_load_b128 in flight per wave before the
    // dependent ds_add_u32 chains -> better L2/HBM latency tolerance.
    for (; i + stride < nchunks; i += 2 * stride) {
        // gfx1250: global_prefetch_b8 (speculative, OOB-safe per ISA)
        __builtin_prefetch((const void*)(p4 + i + 2 * stride), 0, 0);
        __builtin_prefetch((const void*)(t4 + i + 2 * stride), 0, 0);

        int4 p0 = p4[i];
        int4 t0 = t4[i];
        int4 p1 = p4[i + stride];
        int4 t1 = t4[i + stride];

        // ds_add_u32 per element; mismatches skipped (reference dumps them
        // into bin 0, which is excluded from the Dice mean).
        atomicAdd(&hp[p0.x], 1u); atomicAdd(&ht[t0.x], 1u); if (p0.x == t0.x) atomicAdd(&hi[p0.x], 1u);
        atomicAdd(&hp[p0.y], 1u); atomicAdd(&ht[t0.y], 1u); if (p0.y == t0.y) atomicAdd(&hi[p0.y], 1u);
        atomicAdd(&hp[p0.z], 1u); atomicAdd(&ht[t0.z], 1u); if (p0.z == t0.z) atomicAdd(&hi[p0.z], 1u);
        atomicAdd(&hp[p0.w], 1u); atomicAdd(&ht[t0.w], 1u); if (p0.w == t0.w) atomicAdd(&hi[p0.w], 1u);

        atomicAdd(&hp[p1.x], 1u); atomicAdd(&ht[t1.x], 1u); if (p1.x == t1.x) atomicAdd(&hi[p1.x], 1u);
        atomicAdd(&hp[p1.y], 1u); atomicAdd(&ht[t1.y], 1u); if (p1.y == t1.y) atomicAdd(&hi[p1.y], 1u);
        atomicAdd(&hp[p1.z], 1u); atomicAdd(&ht[t1.z], 1u); if (p1.z == t1.z) atomicAdd(&hi[p1.z], 1u);
        atomicAdd(&hp[p1.w], 1u); atomicAdd(&ht[t1.w], 1u); if (p1.w == t1.w) atomicAdd(&hi[p1.w], 1u);
    }
    // Tail (not taken for 256^3, kept for generality).
    for (; i < nchunks; i += stride) {
        int4 p = p4[i];
        int4 t = t4[i];
        atomicAdd(&hp[p.x], 1u); atomicAdd(&ht[t.x], 1u); if (p.x == t.x) atomicAdd(&hi[p.x], 1u);
        atomicAdd(&hp[p.y], 1u); atomicAdd(&ht[t.y], 1u); if (p.y == t.y) atomicAdd(&hi[p.y], 1u);
        atomicAdd(&hp[p.z], 1u); atomicAdd(&ht[t.z], 1u); if (p.z == t.z) atomicAdd(&hi[p.z], 1u);
        atomicAdd(&hp[p.w], 1u); atomicAdd(&ht[t.w], 1u); if (p.w == t.w) atomicAdd(&hi[p.w], 1u);
    }
    __syncthreads();

    // Merge 8 wave-private copies, then one global atomic per nonzero counter.
    for (int c = threadIdx.x; c < 3 * NBINS; c += THREADS) {
        unsigned s = 0;
        #pragma unroll
        for (int w = 0; w < WAVES; ++w) s += lh[w][c];
        if (s) atomicAdd(&ghist[c], s);         // global_atomic_add_u32
    }
}

// ---------------------------------------------------------------- finalize
__global__ void finalize_kernel(const unsigned* __restrict__ ghist,
                                float* __restrict__ out) {
    __shared__ float d[NBINS];
    int b = threadIdx.x;                        // one block of 64 threads
    if (b < NBINS) {
        float v = 0.0f;
        if (b >= 1 && b < 57) {
            float inter = (float)ghist[2 * NBINS + b];
            float uni   = (float)(ghist[b] + ghist[NBINS + b]);
            v = 2.0f * inter / (uni + 1e-5f);
        }
        d[b] = v;
    }
    __syncthreads();
    if (b == 0) {
        // Fixed-order serial sum -> bit-deterministic across replays.
        float s = 0.0f;
        for (int i = 1; i < 57; ++i) s += d[i];
        out[0] = 1.0f - s / 56.0f;
    }
}

// ---------------------------------------------------------------- launch
extern "C" void kernel_launch(void* const* d_in, const int* in_sizes, int n_in,
                              void* d_out, int out_size, void* d_ws, size_t ws_size,
                              hipStream_t stream) {
    const int* pred = (const int*)d_in[0];
    const int* tgt  = (const int*)d_in[1];
    unsigned*  ghist = (unsigned*)d_ws;         // 3*64 u32 = 768 B scratch
    float*     out   = (float*)d_out;

    long long n       = (long long)in_sizes[0]; // 256^3, divisible by 4
    long long nchunks = n >> 2;                 // int4 chunks

    zero_hist_kernel<<<1, 256, 0, stream>>>(ghist);
    hist_kernel<<<GRID, THREADS, 0, stream>>>(pred, tgt, ghist, nchunks);
    finalize_kernel<<<1, 64, 0, stream>>>(ghist, out);
}